// GAT_33260226740749
// MI455X (gfx1250) — compile-verified
//
#include <hip/hip_runtime.h>
#include <hip/hip_bf16.h>

typedef __attribute__((ext_vector_type(2))) float v2f;
typedef __attribute__((ext_vector_type(8))) float v8f;

#define N_NODES   50000
#define N_EDGES   800000
#define EP        (N_EDGES + N_NODES)     // edges incl. self loops
#define N_GRAPHS  512
#define HC        128                     // H*C
#define NHEAD     8
#define NEG_SLOPE 0.2f

// ---------------- order-preserving float<->uint keys for atomic max ----------
__device__ __forceinline__ unsigned fkey(float f) {
    unsigned b = __float_as_uint(f);
    return (b & 0x80000000u) ? ~b : (b | 0x80000000u);
}
__device__ __forceinline__ float funkey(unsigned k) {
    unsigned b = (k & 0x80000000u) ? (k & 0x7FFFFFFFu) : ~k;
    return __uint_as_float(b);
}

// ---------------- self-loop edge_attr = mean of incoming edge attrs ----------
__global__ void gat_loop_sum(const int* __restrict__ ei, const float* __restrict__ ea,
                             float* __restrict__ loop, float* __restrict__ cnt) {
    int e = blockIdx.x * blockDim.x + threadIdx.x;
    if (e >= N_EDGES) return;
    int t = ei[N_EDGES + e];
#pragma unroll
    for (int k = 0; k < 4; ++k) atomicAdd(&loop[t * 4 + k], ea[e * 4 + k]);
    atomicAdd(&cnt[t], 1.0f);
}

__global__ void gat_loop_div(float* __restrict__ loop, const float* __restrict__ cnt) {
    int i = blockIdx.x * blockDim.x + threadIdx.x;
    if (i >= N_NODES * 4) return;
    loop[i] /= fmaxf(cnt[i >> 2], 1.0f);
}

// ---------------- dense transform: [N,4] @ [4,128] via one f32 WMMA ----------
__global__ void gat_gemm_k4(const float* __restrict__ A, const float* __restrict__ B,
                            float* __restrict__ C) {
    int gtid = blockIdx.x * blockDim.x + threadIdx.x;
    int wave = gtid >> 5;
    int lane = gtid & 31;
    int rt = wave >> 3;           // row tile (16 rows)
    int ct = wave & 7;            // col tile (16 cols of 128)
    if (rt >= N_NODES / 16) return;
    int half = lane >> 4, l = lane & 15;
    int k0 = half * 2;
    int row = rt * 16 + l;
    int col = ct * 16 + l;
    // A 16x4 f32 fragment: lanes 0-15 -> K=0,1 ; lanes 16-31 -> K=2,3
    v2f a; a.x = A[row * 4 + k0]; a.y = A[row * 4 + k0 + 1];
    // B 4x16 f32 fragment (mirrored): lanes 0-15 rows 0,1 ; lanes 16-31 rows 2,3
    v2f b; b.x = B[k0 * HC + col]; b.y = B[(k0 + 1) * HC + col];
    v8f c = {};
    c = __builtin_amdgcn_wmma_f32_16x16x4_f32(false, a, false, b, (short)0, c, false, false);
#pragma unroll
    for (int r = 0; r < 8; ++r) {
        int m = rt * 16 + half * 8 + r;   // D layout: M = r + 8*half, N = l
        C[(size_t)m * HC + col] = c[r];
    }
}

// ---------------- dense transform: [N,128] @ [128,128] ------------------------
// W2 (64 KB) staged once per block into the 320 KB WGP LDS; each of the 8 waves
// owns a full 16-row x 128-col strip: A fragments loaded once per k-step and
// reused across all 8 column tiles (8 WMMAs/k-step, f32 accumulation).
__global__ void __launch_bounds__(256) gat_gemm_k128(const float* __restrict__ A,
                                                     const float* __restrict__ B,
                                                     float* __restrict__ C) {
    __shared__ float Bs[HC * HC];                 // 128x128 f32 = 64 KB
    {
        const float4* Bv  = (const float4*)B;
        float4*       Bsv = (float4*)Bs;
        for (int idx = threadIdx.x; idx < HC * HC / 4; idx += 256) Bsv[idx] = Bv[idx];
    }
    __syncthreads();

    int wave = blockIdx.x * 8 + (threadIdx.x >> 5);   // one 16-row tile per wave
    int lane = threadIdx.x & 31;
    if (wave >= N_NODES / 16) return;
    int half = lane >> 4, l = lane & 15;
    int k0 = half * 2;
    int row = wave * 16 + l;

    v8f acc[8] = {};
    for (int kt = 0; kt < 32; ++kt) {
        int kb = kt * 4 + k0;
        v2f a; a.x = A[(size_t)row * HC + kb]; a.y = A[(size_t)row * HC + kb + 1];
#pragma unroll
        for (int ct = 0; ct < 8; ++ct) {
            int col = ct * 16 + l;
            v2f b; b.x = Bs[kb * HC + col]; b.y = Bs[(kb + 1) * HC + col];
            acc[ct] = __builtin_amdgcn_wmma_f32_16x16x4_f32(false, a, false, b,
                                                            (short)0, acc[ct], false, false);
        }
    }
#pragma unroll
    for (int ct = 0; ct < 8; ++ct) {
        int col = ct * 16 + l;
#pragma unroll
        for (int r = 0; r < 8; ++r) {
            int m = wave * 16 + half * 8 + r;     // D layout: M = r + 8*half, N = l
            C[(size_t)m * HC + col] = acc[ct][r];
        }
    }
}

// ---------------- per-node attention projections a_s, a_d --------------------
__global__ void gat_asd(const float* __restrict__ h, const float* __restrict__ att_s,
                        const float* __restrict__ att_d,
                        float* __restrict__ as_, float* __restrict__ ad_) {
    int i = blockIdx.x * blockDim.x + threadIdx.x;
    if (i >= N_NODES * NHEAD) return;
    int n = i >> 3, hd = i & 7;
    const float* hp = h + (size_t)n * HC + hd * 16;
    float s = 0.f, d = 0.f;
#pragma unroll
    for (int c = 0; c < 16; ++c) {
        float v = hp[c];
        s += v * att_s[hd * 16 + c];
        d += v * att_d[hd * 16 + c];
    }
    as_[i] = s; ad_[i] = d;
}

// ---------------- fold (edge_attr@We)*att_edge into 4x8 matrix M -------------
__global__ void gat_attM(const float* __restrict__ We, const float* __restrict__ ae,
                         float* __restrict__ M) {
    int t = threadIdx.x;
    if (t >= 32) return;
    int k = t >> 3, hd = t & 7;
    float m = 0.f;
#pragma unroll
    for (int c = 0; c < 16; ++c) m += We[k * HC + hd * 16 + c] * ae[hd * 16 + c];
    M[t] = m;   // M[k*8+hd]
}

// ---------------- pass A: logits + segment max -------------------------------
__global__ void gat_alpha(const int* __restrict__ ei, const float* __restrict__ ea,
                          const float* __restrict__ loop,
                          const float* __restrict__ as_, const float* __restrict__ ad_,
                          const float* __restrict__ M,
                          float* __restrict__ alpha, unsigned* __restrict__ amax) {
    int i = blockIdx.x * blockDim.x + threadIdx.x;
    if (i >= EP * NHEAD) return;
    int e = i >> 3, hd = i & 7;
    int s, t; const float* eap;
    if (e < N_EDGES) { s = ei[e]; t = ei[N_EDGES + e]; eap = ea + (size_t)e * 4; }
    else             { s = t = e - N_EDGES;            eap = loop + (size_t)(e - N_EDGES) * 4; }
    float aev = 0.f;
#pragma unroll
    for (int k = 0; k < 4; ++k) aev += eap[k] * M[k * 8 + hd];
    float a = as_[s * 8 + hd] + ad_[t * 8 + hd] + aev;
    a = (a > 0.f) ? a : NEG_SLOPE * a;          // leaky_relu
    alpha[i] = a;
    atomicMax(&amax[t * 8 + hd], fkey(a));
}

// ---------------- decode amax keys (non-finite -> 0, matches reference) ------
__global__ void gat_decode(unsigned* __restrict__ amax) {
    int i = blockIdx.x * blockDim.x + threadIdx.x;
    if (i >= N_NODES * NHEAD) return;
    float f = funkey(amax[i]);
    if (!__builtin_isfinite(f)) f = 0.0f;
    amax[i] = __float_as_uint(f);
}

// ---------------- pass B: exp + segment sum ----------------------------------
__global__ void gat_exps(const int* __restrict__ ei, float* __restrict__ alpha,
                         const float* __restrict__ amaxf, float* __restrict__ den) {
    int i = blockIdx.x * blockDim.x + threadIdx.x;
    if (i >= EP * NHEAD) return;
    int e = i >> 3, hd = i & 7;
    int t = (e < N_EDGES) ? ei[N_EDGES + e] : (e - N_EDGES);
    float ex = __expf(alpha[i] - amaxf[t * 8 + hd]);
    alpha[i] = ex;
    atomicAdd(&den[t * 8 + hd], ex);
}

// ---------------- pass C: weighted scatter-add (1 wave per edge) -------------
__global__ void gat_scatter(const int* __restrict__ ei, const float* __restrict__ hin,
                            const float* __restrict__ ex, const float* __restrict__ den,
                            float* __restrict__ out) {
    int i = blockIdx.x * blockDim.x + threadIdx.x;
    if (i >= EP * 32) return;
    int e = i >> 5, lane = i & 31;
    int s, t;
    if (e < N_EDGES) { s = ei[e]; t = ei[N_EDGES + e]; }
    else             { s = t = e - N_EDGES; }
    // overlap random-row fetch with the weight computation (global_prefetch_b8)
    const float* hp = hin + (size_t)s * HC + lane * 4;
    __builtin_prefetch(hp, 0, 3);
    int hd = lane >> 2;                          // 4 channels per lane -> head = lane/4
    float w = ex[(size_t)e * 8 + hd] / (den[t * 8 + hd] + 1e-16f);
    const float4 hv = *(const float4*)hp;
    float* op = out + (size_t)t * HC + lane * 4;
    atomicAdd(op + 0, hv.x * w);
    atomicAdd(op + 1, hv.y * w);
    atomicAdd(op + 2, hv.z * w);
    atomicAdd(op + 3, hv.w * w);
}

// ---------------- bias + relu (in place) -------------------------------------
__global__ void gat_bias_relu(float* __restrict__ acc, const float* __restrict__ bias) {
    int i = blockIdx.x * blockDim.x + threadIdx.x;
    if (i >= N_NODES * HC) return;
    float v = acc[i] + bias[i & (HC - 1)];
    acc[i] = fmaxf(v, 0.0f);
}

// ---------------- global mean pool (sums + counts) ---------------------------
__global__ void gat_pool(const int* __restrict__ batch, const float* __restrict__ h,
                         float* __restrict__ pooled, float* __restrict__ gcnt) {
    int i = blockIdx.x * blockDim.x + threadIdx.x;
    if (i >= N_NODES * 32) return;
    int n = i >> 5, lane = i & 31;
    int g = batch[n];
    const float4 hv = *(const float4*)(h + (size_t)n * HC + lane * 4);
    float* pp = pooled + (size_t)g * HC + lane * 4;
    atomicAdd(pp + 0, hv.x);
    atomicAdd(pp + 1, hv.y);
    atomicAdd(pp + 2, hv.z);
    atomicAdd(pp + 3, hv.w);
    if (lane == 0) atomicAdd(&gcnt[g], 1.0f);
}

// ---------------- final FC head ----------------------------------------------
__global__ void gat_fc(const float* __restrict__ pooled, const float* __restrict__ gcnt,
                       const float* __restrict__ fw, const float* __restrict__ fb,
                       float* __restrict__ out) {
    int i = blockIdx.x * blockDim.x + threadIdx.x;
    if (i >= N_GRAPHS * 2) return;
    int g = i >> 1, o = i & 1;
    float inv = 1.0f / fmaxf(gcnt[g], 1.0f);
    float s = fb[o];
    for (int c = 0; c < HC; ++c) s += pooled[(size_t)g * HC + c] * inv * fw[c * 2 + o];
    out[g * 2 + o] = s;
}

// ================= host side ================================================
extern "C" void kernel_launch(void* const* d_in, const int* in_sizes, int n_in,
                              void* d_out, int out_size, void* d_ws, size_t ws_size,
                              hipStream_t stream) {
    const float* x   = (const float*)d_in[0];
    const int*   ei  = (const int*)d_in[1];
    const int*   bat = (const int*)d_in[2];
    const float* ea  = (const float*)d_in[3];
    const float* W1  = (const float*)d_in[4];
    const float* as1 = (const float*)d_in[5];
    const float* ad1 = (const float*)d_in[6];
    const float* We1 = (const float*)d_in[7];
    const float* ae1 = (const float*)d_in[8];
    const float* b1  = (const float*)d_in[9];
    const float* W2  = (const float*)d_in[10];
    const float* as2 = (const float*)d_in[11];
    const float* ad2 = (const float*)d_in[12];
    const float* We2 = (const float*)d_in[13];
    const float* ae2 = (const float*)d_in[14];
    const float* b2  = (const float*)d_in[15];
    const float* fw  = (const float*)d_in[16];
    const float* fb  = (const float*)d_in[17];
    float* out = (float*)d_out;

    float* ws = (float*)d_ws;
    // workspace layout (floats)
    constexpr size_t OFF_LOOP = 0;                                   // N*4
    constexpr size_t OFF_CNT  = OFF_LOOP + (size_t)N_NODES * 4;      // N
    constexpr size_t OFF_HA   = OFF_CNT + N_NODES;                   // N*128
    constexpr size_t OFF_HB   = OFF_HA + (size_t)N_NODES * HC;       // N*128
    constexpr size_t OFF_AS   = OFF_HB + (size_t)N_NODES * HC;       // N*8
    constexpr size_t OFF_AD   = OFF_AS + (size_t)N_NODES * NHEAD;    // N*8
    constexpr size_t OFF_AMAX = OFF_AD + (size_t)N_NODES * NHEAD;    // N*8
    constexpr size_t OFF_DEN  = OFF_AMAX + (size_t)N_NODES * NHEAD;  // N*8
    constexpr size_t OFF_EX   = OFF_DEN + (size_t)N_NODES * NHEAD;   // EP*8
    constexpr size_t OFF_M    = OFF_EX + (size_t)EP * NHEAD;         // 32
    constexpr size_t OFF_POOL = OFF_M + 32;                          // 512*128
    constexpr size_t OFF_GCNT = OFF_POOL + (size_t)N_GRAPHS * HC;    // 512

    float* loop  = ws + OFF_LOOP;
    float* cnt   = ws + OFF_CNT;
    float* hA    = ws + OFF_HA;
    float* hB    = ws + OFF_HB;
    float* aS    = ws + OFF_AS;
    float* aD    = ws + OFF_AD;
    float* amax  = ws + OFF_AMAX;
    float* den   = ws + OFF_DEN;
    float* exb   = ws + OFF_EX;
    float* Mb    = ws + OFF_M;
    float* pool  = ws + OFF_POOL;
    float* gcnt  = ws + OFF_GCNT;

    const int B = 256;
    const int gEdge   = (N_EDGES + B - 1) / B;
    const int gNode4  = (N_NODES * 4 + B - 1) / B;
    const int gNH     = (N_NODES * NHEAD + B - 1) / B;
    const int gGemm4  = (N_NODES / 16) * 8 * 32 / B;         // 3125 blocks
    const int gGemm128 = (N_NODES / 16 + 7) / 8;             // 391 blocks (8 waves each)
    const int gEH     = (EP * NHEAD + B - 1) / B;
    const int gScat   = (EP * 32 + B - 1) / B;
    const int gNHC    = (N_NODES * HC + B - 1) / B;
    const int gPool   = (N_NODES * 32 + B - 1) / B;

    // self-loop edge attrs
    hipMemsetAsync(loop, 0, (size_t)(N_NODES * 4 + N_NODES) * sizeof(float), stream);
    gat_loop_sum<<<gEdge, B, 0, stream>>>(ei, ea, loop, cnt);
    gat_loop_div<<<gNode4, B, 0, stream>>>(loop, cnt);

    // -------- layer 1 --------
    gat_gemm_k4<<<gGemm4, B, 0, stream>>>(x, W1, hA);
    gat_asd<<<gNH, B, 0, stream>>>(hA, as1, ad1, aS, aD);
    gat_attM<<<1, 32, 0, stream>>>(We1, ae1, Mb);
    hipMemsetAsync(amax, 0, (size_t)N_NODES * NHEAD * 2 * sizeof(float), stream); // amax+den
    gat_alpha<<<gEH, B, 0, stream>>>(ei, ea, loop, aS, aD, Mb, exb, (unsigned*)amax);
    gat_decode<<<gNH, B, 0, stream>>>((unsigned*)amax);
    gat_exps<<<gEH, B, 0, stream>>>(ei, exb, amax, den);
    hipMemsetAsync(hB, 0, (size_t)N_NODES * HC * sizeof(float), stream);
    gat_scatter<<<gScat, B, 0, stream>>>(ei, hA, exb, den, hB);
    gat_bias_relu<<<gNHC, B, 0, stream>>>(hB, b1);

    // -------- layer 2 --------
    gat_gemm_k128<<<gGemm128, B, 0, stream>>>(hB, W2, hA);
    gat_asd<<<gNH, B, 0, stream>>>(hA, as2, ad2, aS, aD);
    gat_attM<<<1, 32, 0, stream>>>(We2, ae2, Mb);
    hipMemsetAsync(amax, 0, (size_t)N_NODES * NHEAD * 2 * sizeof(float), stream);
    gat_alpha<<<gEH, B, 0, stream>>>(ei, ea, loop, aS, aD, Mb, exb, (unsigned*)amax);
    gat_decode<<<gNH, B, 0, stream>>>((unsigned*)amax);
    gat_exps<<<gEH, B, 0, stream>>>(ei, exb, amax, den);
    hipMemsetAsync(hB, 0, (size_t)N_NODES * HC * sizeof(float), stream);
    gat_scatter<<<gScat, B, 0, stream>>>(ei, hA, exb, den, hB);
    gat_bias_relu<<<gNHC, B, 0, stream>>>(hB, b2);

    // -------- pool + head --------
    hipMemsetAsync(pool, 0, (size_t)(N_GRAPHS * HC + N_GRAPHS) * sizeof(float), stream);
    gat_pool<<<gPool, B, 0, stream>>>(bat, hB, pool, gcnt);
    gat_fc<<<(N_GRAPHS * 2 + B - 1) / B, B, 0, stream>>>(pool, gcnt, fw, fb, out);

    (void)in_sizes; (void)n_in; (void)out_size; (void)ws_size;
}